// EdgeGCNModel_pos_83141976916260
// MI455X (gfx1250) — compile-verified
//
#include <hip/hip_runtime.h>
#include <math.h>

// Problem constants (match reference)
#define NN   2560          // nodes
#define NE   163840        // edges
#define NH   4             // heads
#define CC   128           // channels per head
#define FF   512           // NH*CC
#define NOE  40            // out edge dim
#define TPB  256

typedef float v2f __attribute__((ext_vector_type(2)));
typedef float v8f __attribute__((ext_vector_type(8)));

#if defined(__HIP_DEVICE_COMPILE__) && __has_builtin(__builtin_amdgcn_wmma_f32_16x16x4_f32)
#define USE_WMMA_F32 1
#else
#define USE_WMMA_F32 0
#endif

// ---------------- wave helpers (wave32) ----------------
__device__ __forceinline__ float wred_sum(float v) {
  #pragma unroll
  for (int m = 16; m > 0; m >>= 1) v += __shfl_xor(v, m, 32);
  return v;
}
__device__ __forceinline__ float wred_max(float v) {
  #pragma unroll
  for (int m = 16; m > 0; m >>= 1) v = fmaxf(v, __shfl_xor(v, m, 32));
  return v;
}

// ---------------- fp32 WMMA 16x16x4 ----------------
// A 16x4: lanes 0-15 rows, vgpr0/1 = k0+2*half / k0+2*half+1
// B 4x16: lanes 0-15 cols, same k striping
// D 16x16: vgpr i -> row i + 8*half, col = lane&15
__device__ __forceinline__ v8f wmma4_f32(v2f a, v2f b, v8f c) {
#if USE_WMMA_F32
  return __builtin_amdgcn_wmma_f32_16x16x4_f32(false, a, false, b, (short)0, c,
                                               false, false);
#else
  int lane = threadIdx.x & 31;
  int lid = lane & 15, half = lane >> 4;
  #pragma unroll
  for (int i = 0; i < 8; ++i) {
    int m = i + (half << 3);
    float d = c[i];
    #pragma unroll
    for (int k = 0; k < 4; ++k) {
      float av = __shfl((k & 1) ? a.y : a.x, m + ((k >> 1) << 4), 32);
      float bv = __shfl((k & 1) ? b.y : b.x, lid + ((k >> 1) << 4), 32);
      d = fmaf(av, bv, d);
    }
    c[i] = d;
  }
  return c;
#endif
}

// ---------------- kernels ----------------

// edge weights: ew = 1/(||pos[src]-pos[dst]|| + 1e-6)
__global__ void k_edge_w(const int* __restrict__ ei, const float* __restrict__ pos,
                         float* __restrict__ ew) {
  int e = blockIdx.x * blockDim.x + threadIdx.x;
  if (e >= NE) return;
  int s = ei[e], d = ei[NE + e];
  float dx = pos[2 * s] - pos[2 * d];
  float dy = pos[2 * s + 1] - pos[2 * d + 1];
  ew[e] = 1.0f / (sqrtf(dx * dx + dy * dy) + 1e-6f);
}

// coef[h] = dot(We[h*CC..], att_e[h*CC..]) ; 1 block, 4 waves
__global__ void k_ecoef(const float* __restrict__ We, const float* __restrict__ ae,
                        float* __restrict__ coef) {
  int head = threadIdx.x >> 5, lane = threadIdx.x & 31;
  const float4* wp = (const float4*)(We + head * CC);
  const float4* ap = (const float4*)(ae + head * CC);
  float4 w = wp[lane], a = ap[lane];
  float v = w.x * a.x + w.y * a.y + w.z * a.z + w.w * a.w;
  v = wred_sum(v);
  if (lane == 0) coef[head] = v;
}

// degree count
__global__ void k_deg(const int* __restrict__ ei, int* __restrict__ deg) {
  int e = blockIdx.x * blockDim.x + threadIdx.x;
  if (e >= NE) return;
  atomicAdd(&deg[ei[NE + e]], 1);
}

// 1-block exclusive scan of deg -> row_ptr (N+1) and cursor (N). 512 thr x 5
__global__ void k_scan(const int* __restrict__ deg, int* __restrict__ row_ptr,
                       int* __restrict__ cursor) {
  __shared__ int s[512];
  int t = threadIdx.x;
  int loc[5];
  int sum = 0;
  #pragma unroll
  for (int j = 0; j < 5; ++j) { loc[j] = sum; sum += deg[t * 5 + j]; }
  s[t] = sum;
  __syncthreads();
  for (int off = 1; off < 512; off <<= 1) {
    int v = (t >= off) ? s[t - off] : 0;
    __syncthreads();
    s[t] += v;
    __syncthreads();
  }
  int base = (t == 0) ? 0 : s[t - 1];
  #pragma unroll
  for (int j = 0; j < 5; ++j) {
    row_ptr[t * 5 + j] = base + loc[j];
    cursor[t * 5 + j] = base + loc[j];
  }
  if (t == 511) row_ptr[NN] = s[511];
}

// CSR fill: per edge scatter (src, ew) bucketed by dst
__global__ void k_fill(const int* __restrict__ ei, const float* __restrict__ ew,
                       int* __restrict__ cursor, int* __restrict__ csr_src,
                       float* __restrict__ csr_ew) {
  int e = blockIdx.x * blockDim.x + threadIdx.x;
  if (e >= NE) return;
  int d = ei[NE + e];
  int p = atomicAdd(&cursor[d], 1);
  csr_src[p] = ei[e];
  csr_ew[p] = ew[e];
}

// C[MxNn] = A[MxK] @ B[KxNn]; one wave computes a 16x64 macro-tile
// (4 WMMA accumulators share the A fragment), with 1-step SW pipeline.
__global__ void k_gemm_wmma(const float* __restrict__ A, const float* __restrict__ B,
                            float* __restrict__ C, int M, int K, int Nn) {
  int wave = (blockIdx.x * blockDim.x + threadIdx.x) >> 5;
  int lane = threadIdx.x & 31;
  int nG = Nn >> 6;                       // groups of 64 cols
  int mTile = wave / nG, nGrp = wave - mTile * nG;
  if (mTile >= (M >> 4)) return;
  int half = lane >> 4, lid = lane & 15;
  int row = (mTile << 4) + lid;
  int col = (nGrp << 6) + lid;            // lanes cover col..col+48 in 4 tiles
  const float* Arow = A + (size_t)row * K + 2 * half;
  const float* Bk0 = B + (size_t)(2 * half) * Nn + col;
  const float* Bk1 = Bk0 + Nn;

  v8f ac0 = {0.f,0.f,0.f,0.f,0.f,0.f,0.f,0.f};
  v8f ac1 = ac0, ac2 = ac0, ac3 = ac0;

  // preload first fragments
  v2f a = *(const v2f*)(Arow);
  v2f b0 = {Bk0[0],  Bk1[0]};
  v2f b1 = {Bk0[16], Bk1[16]};
  v2f b2 = {Bk0[32], Bk1[32]};
  v2f b3 = {Bk0[48], Bk1[48]};

  for (int k0 = 0; k0 < K; k0 += 4) {
    v2f an = a, bn0 = b0, bn1 = b1, bn2 = b2, bn3 = b3;
    if (k0 + 4 < K) {                     // issue next-step loads early
      const float* Ar = Arow + k0 + 4;
      const float* B0 = Bk0 + (size_t)(k0 + 4) * Nn;
      const float* B1 = B0 + Nn;
      an = *(const v2f*)(Ar);
      bn0.x = B0[0];  bn0.y = B1[0];
      bn1.x = B0[16]; bn1.y = B1[16];
      bn2.x = B0[32]; bn2.y = B1[32];
      bn3.x = B0[48]; bn3.y = B1[48];
    }
    ac0 = wmma4_f32(a, b0, ac0);
    ac1 = wmma4_f32(a, b1, ac1);
    ac2 = wmma4_f32(a, b2, ac2);
    ac3 = wmma4_f32(a, b3, ac3);
    a = an; b0 = bn0; b1 = bn1; b2 = bn2; b3 = bn3;
  }
  int mb = mTile << 4;
  #pragma unroll
  for (int i = 0; i < 8; ++i) {
    int r = mb + i + (half << 3);
    float* Cr = C + (size_t)r * Nn + col;
    Cr[0]  = ac0[i];
    Cr[16] = ac1[i];
    Cr[32] = ac2[i];
    Cr[48] = ac3[i];
  }
}

// per node: a_s[n,h], a_d[n,h] dots; one wave per node
__global__ void k_attn(const float* __restrict__ h, const float* __restrict__ att_s,
                       const float* __restrict__ att_d, float* __restrict__ as_o,
                       float* __restrict__ ad_o) {
  int wave = (blockIdx.x * blockDim.x + threadIdx.x) >> 5;
  int lane = threadIdx.x & 31;
  if (wave >= NN) return;
  #pragma unroll
  for (int hd = 0; hd < NH; ++hd) {
    const float4* hp = (const float4*)(h + (size_t)wave * FF + hd * CC);
    const float4* sp = (const float4*)(att_s + hd * CC);
    const float4* dp = (const float4*)(att_d + hd * CC);
    float4 hv = hp[lane], sv = sp[lane], dv = dp[lane];
    float s = hv.x * sv.x + hv.y * sv.y + hv.z * sv.z + hv.w * sv.w;
    float d = hv.x * dv.x + hv.y * dv.y + hv.z * dv.z + hv.w * dv.w;
    s = wred_sum(s);
    d = wred_sum(d);
    if (lane == 0) { as_o[wave * NH + hd] = s; ad_o[wave * NH + hd] = d; }
  }
}

// segment softmax + weighted aggregation; one wave per (node, head)
__global__ void k_agg(const float* __restrict__ h, const float* __restrict__ as_,
                      const float* __restrict__ ad_, const float* __restrict__ coef,
                      const int* __restrict__ row_ptr, const int* __restrict__ csr_src,
                      const float* __restrict__ csr_ew, const float* __restrict__ bias,
                      float* __restrict__ out) {
  int wave = (blockIdx.x * blockDim.x + threadIdx.x) >> 5;
  int lane = threadIdx.x & 31;
  if (wave >= NN * NH) return;
  int n = wave >> 2, hd = wave & 3;
  int b0 = row_ptr[n], b1 = row_ptr[n + 1];
  float adn = ad_[n * NH + hd];
  float cf = coef[hd];
  // pass 1: max logit
  float mx = -3.4e38f;
  for (int i = b0 + lane; i < b1; i += 32) {
    float l = as_[csr_src[i] * NH + hd] + adn + csr_ew[i] * cf;
    l = (l > 0.f) ? l : 0.2f * l;
    mx = fmaxf(mx, l);
  }
  mx = wred_max(mx);
  // pass 2: exp-weighted feature sum (lanes split the 128 channels)
  float4 acc = {0.f, 0.f, 0.f, 0.f};
  float denom = 0.f;
  for (int i = b0; i < b1; ++i) {
    int s = csr_src[i];
    float l = as_[s * NH + hd] + adn + csr_ew[i] * cf;
    l = (l > 0.f) ? l : 0.2f * l;
    float p = __expf(l - mx);
    denom += p;
    float4 hv = ((const float4*)(h + (size_t)s * FF + hd * CC))[lane];
    acc.x = fmaf(p, hv.x, acc.x);
    acc.y = fmaf(p, hv.y, acc.y);
    acc.z = fmaf(p, hv.z, acc.z);
    acc.w = fmaf(p, hv.w, acc.w);
  }
  float inv = 1.0f / (denom + 1e-16f);
  const float4 bv = ((const float4*)(bias + hd * CC))[lane];
  float4 o;
  o.x = acc.x * inv + bv.x;
  o.y = acc.y * inv + bv.y;
  o.z = acc.z * inv + bv.z;
  o.w = acc.w * inv + bv.w;
  ((float4*)(out + (size_t)n * FF + hd * CC))[lane] = o;
}

// BN stats -> per-channel scale/shift; one wave per channel
__global__ void k_bn_stats(const float* __restrict__ x, const float* __restrict__ g,
                           const float* __restrict__ be, float* __restrict__ sc,
                           float* __restrict__ sh) {
  int c = (blockIdx.x * blockDim.x + threadIdx.x) >> 5;
  int lane = threadIdx.x & 31;
  if (c >= FF) return;
  float sum = 0.f, sq = 0.f;
  for (int r = lane; r < NN; r += 32) {
    float v = x[(size_t)r * FF + c];
    sum += v;
    sq = fmaf(v, v, sq);
  }
  sum = wred_sum(sum);
  sq = wred_sum(sq);
  if (lane == 0) {
    float mu = sum * (1.0f / NN);
    float var = sq * (1.0f / NN) - mu * mu;
    float s = g[c] * rsqrtf(var + 1e-5f);
    sc[c] = s;
    sh[c] = be[c] - mu * s;
  }
}

// y = relu(x*scale + shift)
__global__ void k_bn_apply(const float* __restrict__ x, const float* __restrict__ sc,
                           const float* __restrict__ sh, float* __restrict__ y) {
  int i = blockIdx.x * blockDim.x + threadIdx.x;
  if (i >= NN * FF) return;
  int c = i & (FF - 1);
  y[i] = fmaxf(0.f, fmaf(x[i], sc[c], sh[c]));
}

// out[n,40] = (relu'd A @ Wfc + bfc) * mask + adjacency ; WMMA, N padded to 48
__global__ void k_fc_wmma(const float* __restrict__ A, const float* __restrict__ Wfc,
                          const float* __restrict__ bfc, const int* __restrict__ mask,
                          const float* __restrict__ adj, float* __restrict__ out) {
  const int K = FF;
  int wave = (blockIdx.x * blockDim.x + threadIdx.x) >> 5;
  int lane = threadIdx.x & 31;
  const int nT = 3;                                  // 48 padded cols
  int mTile = wave / nT, nTile = wave - mTile * nT;
  if (mTile >= (NN >> 4)) return;
  int half = lane >> 4, lid = lane & 15;
  int row = (mTile << 4) + lid;
  int col = (nTile << 4) + lid;
  bool inb = (col < NOE);
  v8f acc = {0.f, 0.f, 0.f, 0.f, 0.f, 0.f, 0.f, 0.f};
  #pragma unroll 4
  for (int k0 = 0; k0 < K; k0 += 4) {
    int k = k0 + 2 * half;
    v2f a = *(const v2f*)(A + (size_t)row * K + k);
    v2f b = {inb ? Wfc[k * NOE + col] : 0.f,
             inb ? Wfc[(k + 1) * NOE + col] : 0.f};
    acc = wmma4_f32(a, b, acc);
  }
  int mb = mTile << 4;
  if (inb) {
    float bc = bfc[col];
    #pragma unroll
    for (int i = 0; i < 8; ++i) {
      int r = mb + i + (half << 3);
      float v = (acc[i] + bc) * (float)mask[r];
      out[r * NOE + col] = v + adj[r * NOE + col];
    }
  }
}

// ---------------- host ----------------
static inline size_t alnup(size_t x) { return (x + 255) & ~(size_t)255; }

extern "C" void kernel_launch(void* const* d_in, const int* in_sizes, int n_in,
                              void* d_out, int out_size, void* d_ws, size_t ws_size,
                              hipStream_t stream) {
  const float* x      = (const float*)d_in[0];
  const int*   ei     = (const int*)d_in[1];
  const float* pos    = (const float*)d_in[2];
  const int*   mask   = (const int*)d_in[3];
  const float* adj    = (const float*)d_in[4];
  const float* W1     = (const float*)d_in[5];
  const float* asrc1  = (const float*)d_in[6];
  const float* adst1  = (const float*)d_in[7];
  const float* We1    = (const float*)d_in[8];
  const float* ae1    = (const float*)d_in[9];
  const float* bias1  = (const float*)d_in[10];
  const float* g1     = (const float*)d_in[11];
  const float* be1    = (const float*)d_in[12];
  const float* W2     = (const float*)d_in[13];
  const float* asrc2  = (const float*)d_in[14];
  const float* adst2  = (const float*)d_in[15];
  const float* We2    = (const float*)d_in[16];
  const float* ae2    = (const float*)d_in[17];
  const float* bias2  = (const float*)d_in[18];
  const float* g2     = (const float*)d_in[19];
  const float* be2    = (const float*)d_in[20];
  const float* Wfc    = (const float*)d_in[21];
  const float* bfc    = (const float*)d_in[22];
  float* out = (float*)d_out;

  char* ws = (char*)d_ws;
  size_t off = 0;
  auto alloc = [&](size_t bytes) -> void* {
    void* p = ws + off;
    off = alnup(off + bytes);
    return p;
  };
  float* ew      = (float*)alloc(NE * 4);
  float* coef1   = (float*)alloc(64);
  float* coef2   = (float*)alloc(64);
  int*   deg     = (int*)alloc(NN * 4);
  int*   row_ptr = (int*)alloc((NN + 1) * 4);
  int*   cursor  = (int*)alloc(NN * 4);
  int*   csr_src = (int*)alloc(NE * 4);
  float* csr_ew  = (float*)alloc(NE * 4);
  float* as_b    = (float*)alloc(NN * NH * 4);
  float* ad_b    = (float*)alloc(NN * NH * 4);
  float* sc_b    = (float*)alloc(FF * 4);
  float* sh_b    = (float*)alloc(FF * 4);
  float* bufA    = (float*)alloc((size_t)NN * FF * 4);  // h1raw  / e2raw
  float* bufB    = (float*)alloc((size_t)NN * FF * 4);  // h1agg  / e2agg
  float* bufC    = (float*)alloc((size_t)NN * FF * 4);  // h1post / e2post
  (void)ws_size; (void)n_in; (void)in_sizes; (void)out_size;

  const int EB = NE / TPB;                       // 640
  const int GEMM_WAVES = (NN / 16) * (FF / 64);  // 1280 (16x64 per wave)
  const int GEMM_BLK = GEMM_WAVES / 8;           // 160
  const int AGG_BLK = (NN * NH) / 8;             // 1280
  const int ATTN_BLK = NN / 8;                   // 320
  const int BN_BLK = FF / 8;                     // 64
  const int EW_BLK = (NN * FF) / TPB;            // 5120
  const int FC_BLK = ((NN / 16) * 3) / 8;        // 60

  // graph preprocessing (shared by both layers)
  hipMemsetAsync(deg, 0, NN * sizeof(int), stream);
  k_edge_w<<<EB, TPB, 0, stream>>>(ei, pos, ew);
  k_ecoef<<<1, 128, 0, stream>>>(We1, ae1, coef1);
  k_ecoef<<<1, 128, 0, stream>>>(We2, ae2, coef2);
  k_deg<<<EB, TPB, 0, stream>>>(ei, deg);
  k_scan<<<1, 512, 0, stream>>>(deg, row_ptr, cursor);
  k_fill<<<EB, TPB, 0, stream>>>(ei, ew, cursor, csr_src, csr_ew);

  // layer 1
  k_gemm_wmma<<<GEMM_BLK, TPB, 0, stream>>>(x, W1, bufA, NN, 32, FF);
  k_attn<<<ATTN_BLK, TPB, 0, stream>>>(bufA, asrc1, adst1, as_b, ad_b);
  k_agg<<<AGG_BLK, TPB, 0, stream>>>(bufA, as_b, ad_b, coef1, row_ptr, csr_src,
                                     csr_ew, bias1, bufB);
  k_bn_stats<<<BN_BLK, TPB, 0, stream>>>(bufB, g1, be1, sc_b, sh_b);
  k_bn_apply<<<EW_BLK, TPB, 0, stream>>>(bufB, sc_b, sh_b, bufC);

  // layer 2 (aliases layer-1 buffers)
  k_gemm_wmma<<<GEMM_BLK, TPB, 0, stream>>>(bufC, W2, bufA, NN, FF, FF);
  k_attn<<<ATTN_BLK, TPB, 0, stream>>>(bufA, asrc2, adst2, as_b, ad_b);
  k_agg<<<AGG_BLK, TPB, 0, stream>>>(bufA, as_b, ad_b, coef2, row_ptr, csr_src,
                                     csr_ew, bias2, bufB);
  k_bn_stats<<<BN_BLK, TPB, 0, stream>>>(bufB, g2, be2, sc_b, sh_b);
  k_bn_apply<<<EW_BLK, TPB, 0, stream>>>(bufB, sc_b, sh_b, bufC);

  // fused FC + bias + mask + adjacency
  k_fc_wmma<<<FC_BLK, TPB, 0, stream>>>(bufC, Wfc, bfc, mask, adj, out);
}